// RNDModel_27084063768598
// MI455X (gfx1250) — compile-verified
//
#include <hip/hip_runtime.h>

// CDNA5 / gfx1250, wave32. fp32 WMMA 16x16x4 for all dense GEMMs.

typedef float v2f __attribute__((ext_vector_type(2)));
typedef float v8f __attribute__((ext_vector_type(8)));

static inline int cdiv(long a, long b) { return (int)((a + b - 1) / b); }

// ---------------- small utility kernels ----------------

__global__ void k_fill(float* __restrict__ p, float v, long n) {
    long i = (long)blockIdx.x * blockDim.x + threadIdx.x;
    if (i < n) p[i] = v;
}

__global__ void k_deg(const int* __restrict__ col, float* __restrict__ deg, int E) {
    int e = blockIdx.x * blockDim.x + threadIdx.x;
    if (e < E) atomicAdd(&deg[col[e]], 1.0f);
}

__global__ void k_rsqrt(float* __restrict__ d, int n) {
    int i = blockIdx.x * blockDim.x + threadIdx.x;
    if (i < n) d[i] = rsqrtf(d[i]);   // deg >= 1 always (self loop)
}

// out[col[e]] += dis[row]*dis[col] * xw[row]  (per-feature f32 atomics into L2)
__global__ void k_scatter(const int* __restrict__ row, const int* __restrict__ col,
                          const float* __restrict__ dis, const float* __restrict__ xw,
                          float* __restrict__ out, int E, int F) {
    int e = blockIdx.x * blockDim.x + threadIdx.x;
    if (e >= E) return;
    int r = row[e], c = col[e];
    float nrm = dis[r] * dis[c];
    const float* src = xw + (long)r * F;
    float* dst = out + (long)c * F;
    for (int f = 0; f < F; f += 4) {
        float4 v = *(const float4*)(src + f);
        atomicAdd(dst + f + 0, nrm * v.x);
        atomicAdd(dst + f + 1, nrm * v.y);
        atomicAdd(dst + f + 2, nrm * v.z);
        atomicAdd(dst + f + 3, nrm * v.w);
    }
}

// dst[i] = (elu?) (src[i] + b[i%F])
__global__ void k_bias_act(float* __restrict__ dst, const float* __restrict__ src,
                           const float* __restrict__ b, long n, int F, int do_elu) {
    long i = (long)blockIdx.x * blockDim.x + threadIdx.x;
    if (i >= n * (long)F) return;
    float v = src[i] + b[i % F];
    if (do_elu && v <= 0.0f) v = expf(v) - 1.0f;
    dst[i] = v;
}

// ---------------- WMMA GEMM kernels (fp32, 16x16x4) ----------------

// xw = H[n,fin] @ Wpad[fin,16]; only first fout columns stored.
// Epilogue also writes out = dis^2 * xw (self-loop init of the scatter
// accumulator), saving a full extra pass over N*fout.
// One wave per 16-row tile, 8 waves/block. fin in {128,16,8} (multiple of 4).
__global__ void __launch_bounds__(256)
k_gemm_node(const float* __restrict__ H, const float* __restrict__ W,
            const float* __restrict__ dis,
            float* __restrict__ xw, float* __restrict__ out,
            long ntiles, int fin, int fout) {
    __shared__ float Wl[128 * 16];             // fin<=128, padded to 16 cols
    int tid = threadIdx.x;
    for (int idx = tid; idx < fin * 16; idx += blockDim.x) {
        int k = idx >> 4, c = idx & 15;
        Wl[idx] = (c < fout) ? W[k * fout + c] : 0.0f;
    }
    __syncthreads();

    int wave = tid >> 5;
    int lane = tid & 31;
    int half = lane >> 4;                      // 0: lanes 0-15, 1: lanes 16-31
    int m    = lane & 15;                      // row (A) / col (B,C)
    long tile = (long)blockIdx.x * 8 + wave;
    if (tile >= ntiles) return;                // wave-uniform branch
    long row0 = tile * 16;

    const float* Arow = H + (row0 + m) * (long)fin;
    v8f c = {};
    for (int k = 0; k < fin; k += 4) {
        int kk = k + 2 * half;
        v2f a; a.x = Arow[kk];               a.y = Arow[kk + 1];
        v2f b; b.x = Wl[kk * 16 + m];        b.y = Wl[(kk + 1) * 16 + m];
        c = __builtin_amdgcn_wmma_f32_16x16x4_f32(false, a, false, b,
                                                  (short)0, c, false, false);
    }
    if (m < fout) {
        for (int j = 0; j < 8; ++j) {
            long r = row0 + half * 8 + j;
            float v = c[j];
            float d = dis[r];
            xw[r * (long)fout + m]  = v;
            out[r * (long)fout + m] = d * d * v;   // self-loop term
        }
    }
}

// acc[16,Nc] += A[16,K] @ B[K,Nc] over K-chunk blockIdx.y; one wave per 16-wide
// N tile. Partial tiles combined with f32 atomics (acc pre-zeroed).
__global__ void __launch_bounds__(256)
k_gemm_fc(const float* __restrict__ A, const float* __restrict__ B,
          float* __restrict__ acc, long K, int Nc, long kc) {
    int tid = threadIdx.x;
    int wave = tid >> 5;
    int lane = tid & 31;
    int half = lane >> 4;
    int m    = lane & 15;
    int ntile = blockIdx.x * (blockDim.x >> 5) + wave;
    int n = ntile * 16 + m;
    long k0 = (long)blockIdx.y * kc;
    long k1 = k0 + kc; if (k1 > K) k1 = K;     // tail stays multiple of 4

    v8f c = {};
    for (long k = k0; k < k1; k += 4) {
        long kk = k + 2 * half;
        v2f a; a.x = A[(long)m * K + kk];      a.y = A[(long)m * K + kk + 1];
        v2f b; b.x = B[kk * Nc + n];           b.y = B[(kk + 1) * Nc + n];
        c = __builtin_amdgcn_wmma_f32_16x16x4_f32(false, a, false, b,
                                                  (short)0, c, false, false);
    }
    for (int j = 0; j < 8; ++j)
        atomicAdd(&acc[(half * 8 + j) * (long)Nc + n], c[j]);
}

// ---------------- host-side orchestration ----------------

static void gcn_layer(const float* Hin, int fin, int fout,
                      const float* W, const float* b,
                      const int* row, const int* col, const float* dis,
                      float* xwbuf, float* outbuf,
                      long Nn, int E, hipStream_t stream) {
    long ntiles = Nn / 16;
    k_gemm_node<<<cdiv(ntiles, 8), 256, 0, stream>>>(Hin, W, dis, xwbuf, outbuf,
                                                     ntiles, fin, fout);
    long nf = Nn * (long)fout;
    k_scatter<<<cdiv(E, 256), 256, 0, stream>>>(row, col, dis, xwbuf, outbuf, E, fout);
    k_bias_act<<<cdiv(nf, 256), 256, 0, stream>>>(outbuf, outbuf, b, Nn, fout, 1);
}

extern "C" void kernel_launch(void* const* d_in, const int* in_sizes, int n_in,
                              void* d_out, int out_size, void* d_ws, size_t ws_size,
                              hipStream_t stream) {
    const float* x     = (const float*)d_in[0];
    const int*   ei    = (const int*)d_in[1];       // [2,E] int (row; col)
    const float* pW1   = (const float*)d_in[2];  const float* pb1 = (const float*)d_in[3];
    const float* pW2   = (const float*)d_in[4];  const float* pb2 = (const float*)d_in[5];
    const float* pW3   = (const float*)d_in[6];  const float* pb3 = (const float*)d_in[7];
    const float* tW1   = (const float*)d_in[8];  const float* tb1 = (const float*)d_in[9];
    const float* tW2   = (const float*)d_in[10]; const float* tb2 = (const float*)d_in[11];
    const float* tW3   = (const float*)d_in[12]; const float* tb3 = (const float*)d_in[13];
    const float* pfc1W = (const float*)d_in[14]; const float* pfc1b = (const float*)d_in[15];
    const float* pfc2W = (const float*)d_in[16]; const float* pfc2b = (const float*)d_in[17];
    const float* tfc1W = (const float*)d_in[18]; const float* tfc1b = (const float*)d_in[19];

    const long Nn    = in_sizes[0] / 128;           // 160000 nodes
    const int  E     = in_sizes[1] / 2;             // 5.12M edges
    const long fc_in = in_sizes[14] / 256;          // 80000
    const long Bb    = (Nn * 8) / fc_in;            // 16 graphs
    const int* row = ei;
    const int* col = ei + E;

    // workspace layout
    char* ws = (char*)d_ws;
    size_t off = 0;
    auto take = [&](size_t bytes) { char* p = ws + off; off += (bytes + 255) & ~(size_t)255; return p; };
    float* dis  = (float*)take(Nn * sizeof(float));
    float* P0   = (float*)take(Nn * 16 * sizeof(float));
    float* P1   = (float*)take(Nn * 16 * sizeof(float));
    float* P2   = (float*)take(Nn * 16 * sizeof(float));
    float* accA = (float*)take(Bb * 256 * sizeof(float));   // fc1 accum / hidden
    float* accB = (float*)take(Bb * 256 * sizeof(float));   // fc2 accum

    float* outF = (float*)d_out;                    // [Bb*256 pred | Bb*256 targ]

    // ---- shared degree normalization: dis = rsqrt(1 + in-degree) ----
    k_fill<<<cdiv(Nn, 256), 256, 0, stream>>>(dis, 1.0f, Nn);
    k_deg<<<cdiv(E, 256), 256, 0, stream>>>(col, dis, E);
    k_rsqrt<<<cdiv(Nn, 256), 256, 0, stream>>>(dis, (int)Nn);

    const long bn = Bb * 256;
    const long kc = 800;                            // K-chunk (multiple of 4)
    dim3 fcGrid((int)((256 / 16) / 8), cdiv(fc_in, kc));   // (2, 100) -> 1600 waves

    // ---- predictor network ----
    gcn_layer(x,  128, 16, pW1, pb1, row, col, dis, P0, P1, Nn, E, stream);
    gcn_layer(P1,  16,  8, pW2, pb2, row, col, dis, P2, P0, Nn, E, stream);
    gcn_layer(P0,   8,  8, pW3, pb3, row, col, dis, P1, P2, Nn, E, stream);  // h3 in P2

    k_fill<<<cdiv(bn, 256), 256, 0, stream>>>(accA, 0.0f, bn);
    k_gemm_fc<<<fcGrid, 256, 0, stream>>>(P2, pfc1W, accA, fc_in, 256, kc);
    k_bias_act<<<cdiv(bn, 256), 256, 0, stream>>>(accA, accA, pfc1b, Bb, 256, 1);

    k_fill<<<cdiv(bn, 256), 256, 0, stream>>>(accB, 0.0f, bn);
    k_gemm_fc<<<dim3(2, 1), 256, 0, stream>>>(accA, pfc2W, accB, 256, 256, 256);
    k_bias_act<<<cdiv(bn, 256), 256, 0, stream>>>(outF, accB, pfc2b, Bb, 256, 0);

    // ---- target network (frozen) ----
    gcn_layer(x,  128, 16, tW1, tb1, row, col, dis, P0, P1, Nn, E, stream);
    gcn_layer(P1,  16,  8, tW2, tb2, row, col, dis, P2, P0, Nn, E, stream);
    gcn_layer(P0,   8,  8, tW3, tb3, row, col, dis, P1, P2, Nn, E, stream);  // h3 in P2

    k_fill<<<cdiv(bn, 256), 256, 0, stream>>>(accA, 0.0f, bn);
    k_gemm_fc<<<fcGrid, 256, 0, stream>>>(P2, tfc1W, accA, fc_in, 256, kc);
    k_bias_act<<<cdiv(bn, 256), 256, 0, stream>>>(outF + bn, accA, tfc1b, Bb, 256, 0);
}